// MetaNetImageEncoder_67705864454399
// MI455X (gfx1250) — compile-verified
//
#include <hip/hip_runtime.h>
#include <hip/hip_bf16.h>

// ---------------------------------------------------------------------------
// MetaNet image encoder, CDNA5 (gfx1250) implementation.
// Memory-bound (~375 MB f32 streamed, ~3.1 GFLOP): keep everything f32 and
// use V_WMMA_F32_16X16X4_F32 (M=16 == batch == one WMMA tile row).
// ---------------------------------------------------------------------------

typedef __attribute__((ext_vector_type(2))) float v2f;
typedef __attribute__((ext_vector_type(8))) float v8f;

#define B_  16
#define DIN 9408
#define H1  1024
#define F_  512
#define MH  128
#define T_  8

// ---------------------------------------------------------------------------
// Core WMMA K-loop: acc += (scale * reluA?(A[16 x K])) * B[K x 16] over
// k in [k0,k1), A row-major (lda), B row-major (LDB compile-time so the
// per-step B offsets fold into 24-bit instruction immediates), output tile
// cols n0..n0+15.  Fragment layout per CDNA5 ISA 7.12.2:
//   A 16x4 : lane l -> row M = l&15, VGPR v -> K = v + 2*(l>>4)
//   B 4x16 : lane l -> col N = l&15, VGPR v -> K = v + 2*(l>>4)
//   C 16x16: lane l, VGPR v -> (M = v + 8*(l>>4), N = l&15)
// PF: speculative global_prefetch of the weight stream 32 rows (8 iters)
// ahead; OOB prefetches are silently dropped (ISA 10.5).
// ---------------------------------------------------------------------------
template <bool RELU_A, int LDB, bool PF>
__device__ inline v8f wmma_acc_f32(const float* __restrict__ A, int lda,
                                   const float* __restrict__ B, int n0,
                                   int k0, int k1, v8f acc, float ascale) {
  const int l  = threadIdx.x & 31;
  const int m  = l & 15;            // A row / B col owned by this lane
  const int kk = (l >> 4) << 1;     // K sub-offset for upper half-wave
  const float* __restrict__ arow = A + (size_t)m * lda + k0 + kk;
  const float* __restrict__ bp   = B + (size_t)(k0 + kk) * LDB + n0 + m;
  const int steps = (k1 - k0) >> 2;
#pragma unroll 4
  for (int s = 0; s < steps; ++s) {
    float a0 = arow[0];
    float a1 = arow[1];
    if (RELU_A) { a0 = fmaxf(a0, 0.0f); a1 = fmaxf(a1, 0.0f); }
    v2f a, b;
    a.x = a0 * ascale;
    a.y = a1 * ascale;
    b.x = bp[0];
    b.y = bp[LDB];
    if (PF) {
      __builtin_prefetch(bp + 32 * LDB, 0, 3);
      __builtin_prefetch(bp + 33 * LDB, 0, 3);
    }
    acc = __builtin_amdgcn_wmma_f32_16x16x4_f32(
        /*neg_a=*/false, a, /*neg_b=*/false, b,
        /*c_mod=*/(short)0, acc, /*reuse_a=*/false, /*reuse_b=*/false);
    arow += 4;
    bp   += 4 * LDB;
  }
  return acc;
}

// Store this wave's 16x16 f32 accumulator tile into its LDS reduction slot.
__device__ inline void stash_tile(float* red, int wave, v8f acc) {
  const int l  = threadIdx.x & 31;
  const int n  = l & 15;
  const int mb = (l >> 4) << 3;     // 0 or 8
#pragma unroll
  for (int v = 0; v < 8; ++v)
    red[wave * 256 + (mb + v) * 16 + n] = acc[v];
}

// ---------------------------------------------------------------------------
// k1: 4x4 average pool  x(16,3,224,224) -> xp(16,9408)
// ---------------------------------------------------------------------------
__global__ void pool_kernel(const float* __restrict__ x, float* __restrict__ xp) {
  int idx = blockIdx.x * blockDim.x + threadIdx.x;
  if (idx >= B_ * DIN) return;
  int b  = idx / DIN;
  int r  = idx % DIN;              // c*3136 + oh*56 + ow
  int c  = r / 3136;
  int rr = r % 3136;
  int oh = rr / 56, ow = rr % 56;
  const float* src = x + (((size_t)(b * 3 + c) * 224) + oh * 4) * 224 + ow * 4;
  float s = 0.0f;
#pragma unroll
  for (int i = 0; i < 4; ++i) {
    float4 v = *(const float4*)(src + i * 224);
    s += v.x + v.y + v.z + v.w;
  }
  xp[idx] = s * (1.0f / 16.0f);
}

// ---------------------------------------------------------------------------
// k2: z1 = xp @ W1 + b1   (16x1024). 64 blocks (one N tile each),
// 16 waves split K=9408 (588 each), LDS reduce.
// ---------------------------------------------------------------------------
__global__ void z1_kernel(const float* __restrict__ xp, const float* __restrict__ W1,
                          const float* __restrict__ b1, float* __restrict__ z1) {
  __shared__ float red[16 * 256];
  const int n0 = blockIdx.x * 16;
  const int w  = threadIdx.x >> 5;
  v8f acc = {};
  acc = wmma_acc_f32<false, H1, true>(xp, DIN, W1, n0, w * 588, w * 588 + 588, acc, 1.0f);
  stash_tile(red, w, acc);
  __syncthreads();
  if (threadIdx.x < 256) {
    const int e = threadIdx.x;       // 256 output elements of the tile
    float s = 0.0f;
#pragma unroll
    for (int ww = 0; ww < 16; ++ww) s += red[ww * 256 + e];
    const int b = e >> 4, nn = n0 + (e & 15);
    z1[b * H1 + nn] = s + b1[nn];
  }
}

// ---------------------------------------------------------------------------
// k3: coefficient head (single block, 256 threads, LDS-staged intermediates)
//   base  = relu(z1) @ W2 + b2          (16x512)   -> LDS
//   m     = relu(base @ mW1 + mb1)      (16x128)   -> LDS
//   coefs = m @ mW2 + mb2               (16x32)    -> ws
// ---------------------------------------------------------------------------
__global__ void coef_kernel(const float* __restrict__ z1, const float* __restrict__ W2,
                            const float* __restrict__ b2, const float* __restrict__ mW1,
                            const float* __restrict__ mb1, const float* __restrict__ mW2,
                            const float* __restrict__ mb2, float* __restrict__ coefs) {
  __shared__ float sbase[B_ * F_];   // 32 KB
  __shared__ float sm[B_ * MH];      //  8 KB
  const int w  = threadIdx.x >> 5;
  const int l  = threadIdx.x & 31;
  const int n  = l & 15;
  const int mb = (l >> 4) << 3;

  // phase 1: 32 N-tiles of base, 4 per wave, K = 1024, relu applied on A load
  for (int tile = w * 4; tile < w * 4 + 4; ++tile) {
    v8f acc = {};
    acc = wmma_acc_f32<true, F_, false>(z1, H1, W2, tile * 16, 0, H1, acc, 1.0f);
#pragma unroll
    for (int v = 0; v < 8; ++v)
      sbase[(mb + v) * F_ + tile * 16 + n] = acc[v] + b2[tile * 16 + n];
  }
  __syncthreads();

  // phase 2: 8 N-tiles of m, one per wave, K = 512 (A fragments from LDS)
  {
    v8f acc = {};
    acc = wmma_acc_f32<false, MH, false>(sbase, F_, mW1, w * 16, 0, F_, acc, 1.0f);
#pragma unroll
    for (int v = 0; v < 8; ++v)
      sm[(mb + v) * MH + w * 16 + n] = fmaxf(acc[v] + mb1[w * 16 + n], 0.0f);
  }
  __syncthreads();

  // phase 3: 2 N-tiles of coefs, waves 0..1, K = 128
  if (w < 2) {
    v8f acc = {};
    acc = wmma_acc_f32<false, T_ * 4, false>(sm, MH, mW2, w * 16, 0, MH, acc, 1.0f);
#pragma unroll
    for (int v = 0; v < 8; ++v)
      coefs[(mb + v) * (T_ * 4) + w * 16 + n] = acc[v] + mb2[w * 16 + n];
  }
}

// ---------------------------------------------------------------------------
// k4: h = relu(z1 + sum_t cW1[b,t]*(xp @ dW1[t]) + sum_t cb1[b,t]*db1[t])
// The dominant 308 MB dW1 stream. 64 blocks (N tiles), 16 waves:
// wave w -> t = w>>1, half-K slice (w&1)*4704. A fragment pre-scaled by
// cW1[row, t] (constant per lane), LDS reduce, fused bias + relu.
// ---------------------------------------------------------------------------
__global__ void h_kernel(const float* __restrict__ xp, const float* __restrict__ dW1,
                         const float* __restrict__ db1, const float* __restrict__ coefs,
                         const float* __restrict__ z1, float* __restrict__ h) {
  __shared__ float red[16 * 256];
  const int n0 = blockIdx.x * 16;
  const int w  = threadIdx.x >> 5;
  const int t  = w >> 1;
  const int kb = (w & 1) * (DIN / 2);       // 0 or 4704 (both % 4 == 0)
  const int l  = threadIdx.x & 31;
  const float scale = coefs[(l & 15) * (T_ * 4) + t * 4 + 0];   // cW1[M, t]
  v8f acc = {};
  acc = wmma_acc_f32<false, H1, true>(xp, DIN, dW1 + (size_t)t * DIN * H1,
                                      n0, kb, kb + DIN / 2, acc, scale);
  stash_tile(red, w, acc);
  __syncthreads();
  if (threadIdx.x < 256) {
    const int e = threadIdx.x;
    float s = 0.0f;
#pragma unroll
    for (int ww = 0; ww < 16; ++ww) s += red[ww * 256 + e];
    const int b = e >> 4, nn = n0 + (e & 15);
    float extra = 0.0f;
#pragma unroll
    for (int tt = 0; tt < T_; ++tt)
      extra += coefs[b * (T_ * 4) + tt * 4 + 1] * db1[tt * H1 + nn];
    h[b * H1 + nn] = fmaxf(z1[b * H1 + nn] + s + extra, 0.0f);
  }
}

// ---------------------------------------------------------------------------
// k5: out = h @ W2 + sum_t cW2[b,t]*(h @ dW2[t]) + b2 + sum_t cb2[b,t]*db2[t]
// 32 blocks (N tiles), 16 waves; wave w accumulates BOTH the cW2-scaled
// half-K slice of the dW2[t = w>>1] GEMM and its 64-wide K-slice of h @ W2
// into one accumulator; LDS reduce, fused bias terms.
// ---------------------------------------------------------------------------
__global__ void out_kernel(const float* __restrict__ h, const float* __restrict__ W2,
                           const float* __restrict__ b2, const float* __restrict__ dW2,
                           const float* __restrict__ db2, const float* __restrict__ coefs,
                           float* __restrict__ out) {
  __shared__ float red[16 * 256];
  const int n0 = blockIdx.x * 16;
  const int w  = threadIdx.x >> 5;
  const int t  = w >> 1;
  const int kb = (w & 1) * (H1 / 2);        // 0 or 512
  const int l  = threadIdx.x & 31;
  const float scale = coefs[(l & 15) * (T_ * 4) + t * 4 + 2];   // cW2[M, t]
  v8f acc = {};
  acc = wmma_acc_f32<false, F_, true>(h, H1, dW2 + (size_t)t * H1 * F_,
                                      n0, kb, kb + H1 / 2, acc, scale);
  acc = wmma_acc_f32<false, F_, false>(h, H1, W2, n0, w * 64, w * 64 + 64, acc, 1.0f);
  stash_tile(red, w, acc);
  __syncthreads();
  if (threadIdx.x < 256) {
    const int e = threadIdx.x;
    float s = 0.0f;
#pragma unroll
    for (int ww = 0; ww < 16; ++ww) s += red[ww * 256 + e];
    const int b = e >> 4, nn = n0 + (e & 15);
    float extra = b2[nn];
#pragma unroll
    for (int tt = 0; tt < T_; ++tt)
      extra += coefs[b * (T_ * 4) + tt * 4 + 3] * db2[tt * F_ + nn];
    out[b * F_ + nn] = s + extra;
  }
}

// ---------------------------------------------------------------------------
extern "C" void kernel_launch(void* const* d_in, const int* in_sizes, int n_in,
                              void* d_out, int out_size, void* d_ws, size_t ws_size,
                              hipStream_t stream) {
  const float* x   = (const float*)d_in[0];
  const float* W1  = (const float*)d_in[1];
  const float* b1  = (const float*)d_in[2];
  const float* W2  = (const float*)d_in[3];
  const float* b2  = (const float*)d_in[4];
  const float* dW1 = (const float*)d_in[5];
  const float* db1 = (const float*)d_in[6];
  const float* dW2 = (const float*)d_in[7];
  const float* db2 = (const float*)d_in[8];
  const float* mW1 = (const float*)d_in[9];
  const float* mb1 = (const float*)d_in[10];
  const float* mW2 = (const float*)d_in[11];
  const float* mb2 = (const float*)d_in[12];
  float* out = (float*)d_out;

  float* ws    = (float*)d_ws;
  float* xp    = ws;                 // 16*9408 = 150528
  float* z1    = xp + B_ * DIN;      // 16*1024 =  16384
  float* coefs = z1 + B_ * H1;       // 16*32   =    512
  float* h     = coefs + B_ * 32;    // 16*1024 =  16384
  (void)in_sizes; (void)n_in; (void)out_size; (void)ws_size;

  pool_kernel<<<dim3((B_ * DIN + 255) / 256), dim3(256), 0, stream>>>(x, xp);
  z1_kernel  <<<dim3(H1 / 16), dim3(512), 0, stream>>>(xp, W1, b1, z1);
  coef_kernel<<<dim3(1),       dim3(256), 0, stream>>>(z1, W2, b2, mW1, mb1, mW2, mb2, coefs);
  h_kernel   <<<dim3(H1 / 16), dim3(512), 0, stream>>>(xp, dW1, db1, coefs, z1, h);
  out_kernel <<<dim3(F_ / 16), dim3(512), 0, stream>>>(h, W2, b2, dW2, db2, coefs, out);
}